// RecurrentDoubleGNN_6279242187233
// MI455X (gfx1250) — compile-verified
//
#include <hip/hip_runtime.h>
#include <hip/hip_bf16.h>

#define N_NODES 20000
#define N_EDGES 1280000
#define IN_DIM  16
#define HID     128
#define GATES   384
#define OUT_DIM 4

typedef __attribute__((ext_vector_type(2))) float v2f;
typedef __attribute__((ext_vector_type(8))) float v8f;

// ---------------------------------------------------------------- degree prep
__global__ void init_deg_kernel(float* __restrict__ deg, int n) {
    int i = blockIdx.x * blockDim.x + threadIdx.x;
    if (i < n) deg[i] = 1.0f;  // self-loop
}

__global__ void edge_deg_kernel(const int* __restrict__ cols, float* __restrict__ deg, int e) {
    int i = blockIdx.x * blockDim.x + threadIdx.x;
    if (i < e) atomicAdd(&deg[cols[i]], 1.0f);
}

__global__ void dinv_kernel(float* __restrict__ deg, int n) {
    int i = blockIdx.x * blockDim.x + threadIdx.x;
    if (i < n) deg[i] = rsqrtf(deg[i]);  // deg>0 always (self loop)
}

__global__ void zero_kernel(float4* __restrict__ p, int n4) {
    int i = blockIdx.x * blockDim.x + threadIdx.x;
    if (i < n4) p[i] = make_float4(0.f, 0.f, 0.f, 0.f);
}

// ------------------------------------------------- GEMM1: hs = (x @ W_gcn) * dinv[row]
// x:[N,16] W:[16,128].  One wave -> one 16x16 tile, K=16 via 4x V_WMMA_F32_16X16X4_F32.
__global__ __launch_bounds__(256) void gcn_gemm_kernel(
    const float* __restrict__ x, const float* __restrict__ W,
    const float* __restrict__ dinv, float* __restrict__ hs)
{
    const int wave  = threadIdx.x >> 5;       // 0..7 -> ntile
    const int lane  = threadIdx.x & 31;
    const int mtile = blockIdx.x;             // 0..1249
    const int row   = mtile * 16 + (lane & 15);
    const int ncol  = wave * 16 + (lane & 15);
    const int ksel  = (lane >> 4) * 2;        // lanes 16-31 hold K+2,K+3

    v8f c = {};
#pragma unroll
    for (int k0 = 0; k0 < IN_DIM; k0 += 4) {
        const int kb = k0 + ksel;
        v2f a, b;
        a.x = x[row * IN_DIM + kb];
        a.y = x[row * IN_DIM + kb + 1];
        b.x = W[kb * HID + ncol];
        b.y = W[(kb + 1) * HID + ncol];
        c = __builtin_amdgcn_wmma_f32_16x16x4_f32(false, a, false, b, (short)0, c, false, false);
    }
#pragma unroll
    for (int r = 0; r < 8; ++r) {
        const int mrow = mtile * 16 + r + 8 * (lane >> 4);
        hs[mrow * HID + ncol] = c[r] * dinv[mrow];
    }
}

// ------------------------------------------------- edge scatter: accum[col] += hs[row]
// One wave per edge; lane covers 4 floats of the 128-wide row (L2-resident atomics).
__global__ __launch_bounds__(256) void edge_scatter_kernel(
    const int* __restrict__ rows, const int* __restrict__ cols,
    const float* __restrict__ hs, float* __restrict__ accum, int n_edges)
{
    const int wid  = (blockIdx.x * blockDim.x + threadIdx.x) >> 5;
    const int lane = threadIdx.x & 31;
    if (wid >= n_edges) return;
    const int r = rows[wid];
    const int c = cols[wid];
    const float4 v = ((const float4*)(hs + (size_t)r * HID))[lane];
    float* dst = accum + (size_t)c * HID + lane * 4;
    atomicAdd(dst + 0, v.x);
    atomicAdd(dst + 1, v.y);
    atomicAdd(dst + 2, v.z);
    atomicAdd(dst + 3, v.w);
}

// g = dinv[node]*(accum + hs) + b_gcn   (in place over accum)
__global__ void finalize_gcn_kernel(const float* __restrict__ hs, float* __restrict__ accum,
                                    const float* __restrict__ dinv, const float* __restrict__ b,
                                    int total)
{
    int t = blockIdx.x * blockDim.x + threadIdx.x;
    if (t >= total) return;
    const int node = t >> 7;
    const int k = t & (HID - 1);
    accum[t] = dinv[node] * (accum[t] + hs[t]) + b[k];
}

// ------------------------------------------------- GEMM2: xg = g @ W_ih^T + b_ih
// g:[N,128] W_ih:[384,128] row-major -> B[k][n] = W_ih[n*128+k]. K=128 -> 32 WMMAs/tile.
__global__ __launch_bounds__(256) void gate_gemm_kernel(
    const float* __restrict__ g, const float* __restrict__ W_ih,
    const float* __restrict__ b_ih, float* __restrict__ xg)
{
    const int wave  = threadIdx.x >> 5;
    const int lane  = threadIdx.x & 31;
    const int mtile = blockIdx.x / 3;                 // 0..1249
    const int ntile = (blockIdx.x % 3) * 8 + wave;    // 0..23
    const int row   = mtile * 16 + (lane & 15);
    const int ncol  = ntile * 16 + (lane & 15);
    const int ksel  = (lane >> 4) * 2;

    const float2* ga = (const float2*)(g + (size_t)row * HID);
    const float2* gb = (const float2*)(W_ih + (size_t)ncol * HID);

    v8f c = {};
#pragma unroll
    for (int k0 = 0; k0 < HID; k0 += 4) {
        const int kh = (k0 + ksel) >> 1;
        const float2 av = ga[kh];
        const float2 bv = gb[kh];
        v2f a, b;
        a.x = av.x; a.y = av.y;
        b.x = bv.x; b.y = bv.y;
        c = __builtin_amdgcn_wmma_f32_16x16x4_f32(false, a, false, b, (short)0, c, false, false);
    }
    const float bias = b_ih[ncol];
#pragma unroll
    for (int r = 0; r < 8; ++r) {
        const int mrow = mtile * 16 + r + 8 * (lane >> 4);
        xg[(size_t)mrow * GATES + ncol] = c[r] + bias;
    }
}

// ------------------------------------------------- sequential GRU scan (1 workgroup)
// 384 threads: thread j owns W_hh row j in 32 float4 registers (192KB total in VGPRs).
// xg rows are double-buffered into LDS via the CDNA5 async global->LDS DMA path
// (GLOBAL_LOAD_ASYNC_TO_LDS_B32 / ASYNCcnt), issued one step ahead so the global
// load latency overlaps the 128-FMA matvec instead of sitting on the scan's
// critical path.
__global__ __launch_bounds__(384, 1) void gru_scan_kernel(
    const float* __restrict__ xg,    // [N,384] (b_ih folded in)
    const float* __restrict__ W_hh,  // [384,128]
    const float* __restrict__ b_hh,  // [384]
    const float* __restrict__ h0,    // [128]
    float* __restrict__ ys,          // [N,128]
    float* __restrict__ hT_out,      // -> d_out + 160000
    int n_nodes)
{
    __shared__ float h_sh[HID];
    __shared__ float hg_sh[GATES];
    __shared__ float xg_buf[2][GATES];
    const int j = threadIdx.x;

    float4 w[HID / 4];
    const float4* wrow = (const float4*)(W_hh + (size_t)j * HID);
#pragma unroll
    for (int k = 0; k < HID / 4; ++k) w[k] = wrow[k];
    const float bj = b_hh[j];

    if (j < HID) h_sh[j] = h0[j];

    // preload xg row 0 into xg_buf[0] via async DMA (ASYNCcnt)
    {
        const float* gp = xg + j;
        unsigned lofs = (unsigned)(size_t)&xg_buf[0][j];  // addr[31:0] = LDS byte address
        asm volatile("global_load_async_to_lds_b32 %0, %1, off"
                     :: "v"(lofs), "v"(gp) : "memory");
    }
    __syncthreads();

    for (int t = 0; t < n_nodes; ++t) {
        // issue DMA for row t+1 into the other buffer; overlaps the matvec below
        if (t + 1 < n_nodes) {
            const float* gp = xg + (size_t)(t + 1) * GATES + j;
            unsigned lofs = (unsigned)(size_t)&xg_buf[(t + 1) & 1][j];
            asm volatile("global_load_async_to_lds_b32 %0, %1, off"
                         :: "v"(lofs), "v"(gp) : "memory");
        }

        // hg[j] = W_hh[j,:] . h + b_hh[j]   (h broadcast from LDS, weights in VGPRs)
        const float4* h4 = (const float4*)h_sh;
        float acc = bj;
#pragma unroll
        for (int k = 0; k < HID / 4; ++k) {
            const float4 hv = h4[k];
            acc = fmaf(w[k].x, hv.x, acc);
            acc = fmaf(w[k].y, hv.y, acc);
            acc = fmaf(w[k].z, hv.z, acc);
            acc = fmaf(w[k].w, hv.w, acc);
        }
        hg_sh[j] = acc;

        // retire row t's DMA: async loads complete in order, so ASYNCcnt<=1
        // (the newest, row t+1, may still be in flight) implies row t is in LDS.
        if (t + 1 < n_nodes)
            asm volatile("s_wait_asynccnt 0x1" ::: "memory");
        else
            asm volatile("s_wait_asynccnt 0x0" ::: "memory");
        __syncthreads();

        if (j < HID) {
            const float* xr = xg_buf[t & 1];
            const float r = 1.0f / (1.0f + __expf(-(xr[j]           + hg_sh[j])));
            const float z = 1.0f / (1.0f + __expf(-(xr[j + HID]     + hg_sh[j + HID])));
            const float n = tanhf(xr[j + 2 * HID] + r * hg_sh[j + 2 * HID]);
            const float hnew = (1.0f - z) * n + z * h_sh[j];
            h_sh[j] = hnew;
            ys[(size_t)t * HID + j] = hnew;
        }
        __syncthreads();
    }
    if (j < HID) hT_out[j] = h_sh[j];
}

// ------------------------------------------------- readout + new_x assembly
__global__ void readout_kernel(const float* __restrict__ ys, const float* __restrict__ W_fc,
                               const float* __restrict__ b_fc, const float* __restrict__ x,
                               float* __restrict__ out_newx, int n_nodes)
{
    int t = blockIdx.x * blockDim.x + threadIdx.x;
    const int n = t >> 2, o = t & 3;
    if (n >= n_nodes) return;
    float acc = b_fc[o];
    const float* yr = ys + (size_t)n * HID;
#pragma unroll 8
    for (int k = 0; k < HID; ++k) acc = fmaf(yr[k], W_fc[k * OUT_DIM + o], acc);
    float* dst = out_newx + (size_t)n * 8;
    dst[3 + o] = acc;
    if (o == 0) {
        const float* xr = x + (size_t)n * IN_DIM;
        dst[0] = xr[0];
        dst[1] = xr[1];
        dst[2] = xr[2];
        dst[7] = xr[7];
    }
}

// ---------------------------------------------------------------- launcher
extern "C" void kernel_launch(void* const* d_in, const int* in_sizes, int n_in,
                              void* d_out, int out_size, void* d_ws, size_t ws_size,
                              hipStream_t stream) {
    const float* x     = (const float*)d_in[0];
    const int*   ei    = (const int*)d_in[1];
    const float* h0    = (const float*)d_in[2];
    const float* W_gcn = (const float*)d_in[3];
    const float* b_gcn = (const float*)d_in[4];
    const float* W_ih  = (const float*)d_in[5];
    const float* W_hh  = (const float*)d_in[6];
    const float* b_ih  = (const float*)d_in[7];
    const float* b_hh  = (const float*)d_in[8];
    const float* W_fc  = (const float*)d_in[9];
    const float* b_fc  = (const float*)d_in[10];
    float* out = (float*)d_out;

    // workspace layout (floats)
    float* ws    = (float*)d_ws;
    float* dinv  = ws;                       // 20480 (padded)
    float* hs    = dinv + 20480;             // 2,560,000
    float* accum = hs + (size_t)N_NODES*HID; // 2,560,000 (also holds g)
    float* xg    = accum + (size_t)N_NODES*HID; // 7,680,000
    float* ys    = xg + (size_t)N_NODES*GATES;  // 2,560,000

    const int* rows = ei;            // sources
    const int* cols = ei + N_EDGES;  // targets

    // 1) degrees -> dinv
    init_deg_kernel<<<(N_NODES + 255) / 256, 256, 0, stream>>>(dinv, N_NODES);
    edge_deg_kernel<<<(N_EDGES + 255) / 256, 256, 0, stream>>>(cols, dinv, N_EDGES);
    dinv_kernel<<<(N_NODES + 255) / 256, 256, 0, stream>>>(dinv, N_NODES);

    // 2) zero accumulator (ws is poisoned; must re-zero each call)
    {
        const int n4 = (N_NODES * HID) / 4;
        zero_kernel<<<(n4 + 255) / 256, 256, 0, stream>>>((float4*)accum, n4);
    }

    // 3) hs = (x @ W_gcn) * dinv[row]   — WMMA f32
    gcn_gemm_kernel<<<N_NODES / 16, 256, 0, stream>>>(x, W_gcn, dinv, hs);

    // 4) accum[col] += hs[row] over all edges (wave per edge)
    edge_scatter_kernel<<<(N_EDGES * 32) / 256, 256, 0, stream>>>(rows, cols, hs, accum, N_EDGES);

    // 5) g = dinv*(accum + hs) + b_gcn  (self loop folded in)
    finalize_gcn_kernel<<<(N_NODES * HID + 255) / 256, 256, 0, stream>>>(
        hs, accum, dinv, b_gcn, N_NODES * HID);

    // 6) xg = g @ W_ih^T + b_ih  — WMMA f32, K=128
    gate_gemm_kernel<<<(N_NODES / 16) * 3, 256, 0, stream>>>(accum, W_ih, b_ih, xg);

    // 7) sequential GRU scan, hT -> tail of d_out
    gru_scan_kernel<<<1, GATES, 0, stream>>>(xg, W_hh, b_hh, h0, ys,
                                             out + (size_t)N_NODES * 8, N_NODES);

    // 8) readout + assemble new_x
    readout_kernel<<<(N_NODES * OUT_DIM + 255) / 256, 256, 0, stream>>>(
        ys, W_fc, b_fc, x, out, N_NODES);
}